// GeneralizedInteractionNet_87162066305277
// MI455X (gfx1250) — compile-verified
//
#include <hip/hip_runtime.h>
#include <stdint.h>

// GeneralizedInteractionNet on MI455X (gfx1250, wave32).
// B=2048, F=N=40, D=64, L=3.  fp32 end-to-end via V_WMMA_F32_16X16X4_F32,
// LDS filled with GLOBAL_LOAD_ASYNC_TO_LDS_B128 (ASYNCcnt path).

#define BATCH 2048
#define NF    40     // F == N (subspaces/fields)
#define DIM   64     // embed dim
#define LAYERS 3
#define BT    8      // batch tile per block
#define NIPAD 48     // i padded to multiple of 16 for the G GEMM

// LDS layout (floats), dynamic shared:
//   sBi  @ 0            [BT*NF][D]   20480 f  (81920 B)
//   sB0  @ 20480        [BT][NF][D]  20480 f
//   sC   @ 40960        [BT*NF][D]   20480 f
//   sMh  @ 61440        [D][D]        4096 f
//   sAl  @ 65536        [NIPAD][NF]   1920 f
//   sOut @ 67456        [BT][D]        512 f
#define OFF_B0  (BT * NF * DIM)
#define OFF_C   (2 * BT * NF * DIM)
#define OFF_MH  (3 * BT * NF * DIM)
#define OFF_AL  (OFF_MH + DIM * DIM)
#define OFF_OUT (OFF_AL + NIPAD * NF)
#define SMEM_FLOATS (OFF_OUT + BT * DIM)

typedef __attribute__((ext_vector_type(2))) float v2f;
typedef __attribute__((ext_vector_type(8))) float v8f;

// ---- WMMA fragment helpers (CDNA5 layouts, wave32) -------------------------
// A-matrix 16x4 f32 (M x K): lanes 0-15 -> M=lane, regs {K0,K1};
//                            lanes 16-31 -> M=lane-16, regs {K2,K3}.
__device__ __forceinline__ v2f load_afrag(const float* s, int row0, int col0,
                                          int stride, int lane) {
  int m = lane & 15;
  int k = (lane >> 4) << 1;          // 0 or 2
  v2f a;
  a.x = s[(row0 + m) * stride + col0 + k];
  a.y = s[(row0 + m) * stride + col0 + k + 1];
  return a;
}
// B-matrix 4x16 f32 (K x N), element (k, n) = s[(row0+k)*stride + col0 + n]
__device__ __forceinline__ v2f load_bfrag(const float* s, int row0, int col0,
                                          int stride, int lane) {
  int n = lane & 15;
  int k = (lane >> 4) << 1;
  v2f b;
  b.x = s[(row0 + k)     * stride + col0 + n];
  b.y = s[(row0 + k + 1) * stride + col0 + n];
  return b;
}
// B-matrix sourced transposed: element (k, n) = s[(col0+n)*stride + row0 + k]
__device__ __forceinline__ v2f load_bfrag_T(const float* s, int row0, int col0,
                                            int stride, int lane) {
  int n = lane & 15;
  int k = (lane >> 4) << 1;
  v2f b;
  b.x = s[(col0 + n) * stride + row0 + k];
  b.y = s[(col0 + n) * stride + row0 + k + 1];
  return b;
}
// C/D 16x16 f32: reg r, lane l -> row = r + 8*(l>=16), col = l&15
__device__ __forceinline__ void store_cfrag(float* s, const v8f& c, int row0,
                                            int col0, int stride, int lane) {
  int col   = col0 + (lane & 15);
  int rbase = row0 + ((lane >> 4) << 3);
#pragma unroll
  for (int r = 0; r < 8; ++r) s[(rbase + r) * stride + col] = c[r];
}

__device__ __forceinline__ v8f wmma_f32(v2f a, v2f b, v8f c) {
  // (neg_a, A, neg_b, B, c_mod, C, reuse_a, reuse_b)
  return __builtin_amdgcn_wmma_f32_16x16x4_f32(false, a, false, b, (short)0, c,
                                               false, false);
}

// Async global -> LDS, 16B per lane, GVS mode (SGPR base + per-lane i32 off).
__device__ __forceinline__ void async_b128(unsigned lds_byte_off,
                                           unsigned g_byte_off,
                                           const float* base) {
  asm volatile("global_load_async_to_lds_b128 %0, %1, %2"
               :
               : "v"(lds_byte_off), "v"(g_byte_off),
                 "s"((unsigned long long)(uintptr_t)base)
               : "memory");
}
__device__ __forceinline__ void wait_async0() {
  asm volatile("s_wait_asynccnt 0x0" ::: "memory");
}

// ---- prep kernels ----------------------------------------------------------
// Mh[l,n,Dq,d] = W[l,n,Dq,d] * h[l,n,d]
__global__ void prep_mh_kernel(const float* __restrict__ W,
                               const float* __restrict__ h,
                               float* __restrict__ Mh, int total) {
  int idx = blockIdx.x * blockDim.x + threadIdx.x;
  if (idx >= total) return;
  int d  = idx & (DIM - 1);
  int ln = idx >> 12;                 // l*NF + n   (DIM*DIM == 4096)
  Mh[idx] = W[idx] * h[ln * DIM + d];
}
// AlT[l][n][i][f] = (i < NF) ? alpha[l][f][i][n] : 0
__global__ void prep_alt_kernel(const float* __restrict__ alpha,
                                float* __restrict__ AlT, int total) {
  int idx = blockIdx.x * blockDim.x + threadIdx.x;
  if (idx >= total) return;
  int f = idx % NF;
  int i = (idx / NF) % NIPAD;
  int n = (idx / (NF * NIPAD)) % NF;
  int l = idx / (NF * NIPAD * NF);
  AlT[idx] = (i < NF) ? alpha[((l * NF + f) * NF + i) * NF + n] : 0.0f;
}

// ---- fused per-layer kernel ------------------------------------------------
// out[b,n,:] = sum_i G[b,i,:] .* C[b,i,:]
//   C[b,i,:] = Bi[b,i,:] @ Mh[n]^T          (contract d, K=64)
//   G[b,i,:] = alphaT[n][i,:] @ B0[b]       (contract f, K=40)
__global__ void __launch_bounds__(256)
gin_layer_kernel(const float* __restrict__ B0,   // [B, NF, D] layer-0 inputs
                 const float* __restrict__ Bi,   // [B, NF, D] current state
                 const float* __restrict__ Mh,   // [NF, D, D] this layer
                 const float* __restrict__ AlT,  // [NF, NIPAD, NF] this layer
                 float* __restrict__ Out)        // [B, NF, D]
{
  extern __shared__ float smem[];
  float* sBi  = smem;
  float* sB0  = smem + OFF_B0;
  float* sC   = smem + OFF_C;
  float* sMh  = smem + OFF_MH;
  float* sAl  = smem + OFF_AL;
  float* sOut = smem + OFF_OUT;

  const int tid  = threadIdx.x;
  const int lane = tid & 31;
  const int wave = tid >> 5;
  const int b0   = blockIdx.x * BT;

  // Stage the batch tile of Bi and B0 via async global->LDS (ASYNCcnt path).
  {
    const unsigned ldsbase = __builtin_amdgcn_groupstaticsize();
    const float* gBi = Bi + (size_t)b0 * NF * DIM;
    const float* gB0 = B0 + (size_t)b0 * NF * DIM;
    for (int i = tid; i < BT * NF * DIM / 4; i += 256) {
      unsigned off = (unsigned)i * 16u;                 // byte offset in tile
      async_b128(ldsbase + off, off, gBi);              // -> sBi
      async_b128(ldsbase + OFF_B0 * 4u + off, off, gB0);// -> sB0
    }
    wait_async0();
  }
  __syncthreads();

  for (int n = 0; n < NF; ++n) {
    // Stage per-n operands; zero the output accumulator.
    {
      const float4* gM = (const float4*)(Mh + (size_t)n * DIM * DIM);
      float4* sM4 = (float4*)sMh;
      for (int i = tid; i < DIM * DIM / 4; i += 256) sM4[i] = gM[i];
      const float4* gA = (const float4*)(AlT + (size_t)n * NIPAD * NF);
      float4* sA4 = (float4*)sAl;
      for (int i = tid; i < NIPAD * NF / 4; i += 256) sA4[i] = gA[i];
      for (int i = tid; i < BT * DIM; i += 256) sOut[i] = 0.0f;
      if (n + 1 < NF && tid == 0) {         // warm L2/WGP$ for next n
        __builtin_prefetch(Mh  + (size_t)(n + 1) * DIM * DIM, 0, 1);
        __builtin_prefetch(AlT + (size_t)(n + 1) * NIPAD * NF, 0, 1);
      }
    }
    __syncthreads();

    // Stage 1: C = [320 x 64] = sBi(320x64) @ sMh^T(64x64).
    // 20 row-tiles x 2 column-tile pairs = 40 units; A-frag shared by the
    // pair -> half the A-side DS traffic, 2 independent WMMA chains per wave.
    for (int u = wave; u < (BT * NF / 16) * 2; u += 8) {
      int rt = u >> 1, c0 = (u & 1) * 32;   // column base of the pair
      v8f acc0 = {}, acc1 = {};
#pragma unroll
      for (int kk = 0; kk < DIM; kk += 4) {
        v2f a   = load_afrag(sBi, rt * 16, kk, DIM, lane);
        v2f bf0 = load_bfrag_T(sMh, kk, c0,      DIM, lane);  // Mh^T
        v2f bf1 = load_bfrag_T(sMh, kk, c0 + 16, DIM, lane);
        acc0 = wmma_f32(a, bf0, acc0);
        acc1 = wmma_f32(a, bf1, acc1);
      }
      store_cfrag(sC, acc0, rt * 16, c0,      DIM, lane);
      store_cfrag(sC, acc1, rt * 16, c0 + 16, DIM, lane);
    }
    __syncthreads();

    // Stage 2: per b: G = sAl(48x40) @ sB0_b(40x64); one (b, i-tile) unit
    // covers all 4 column tiles sharing one A-frag -> 4 independent chains.
    // Fused with the C elementwise product and i-reduction into sOut.
    for (int u = wave; u < BT * (NIPAD / 16); u += 8) {
      int b = u / 3, it = u % 3;
      v8f acc[4] = {{}, {}, {}, {}};
#pragma unroll
      for (int kk = 0; kk < NF; kk += 4) {
        v2f a = load_afrag(sAl, it * 16, kk, NF, lane);
#pragma unroll
        for (int c = 0; c < 4; ++c) {
          v2f bf = load_bfrag(sB0 + b * NF * DIM, kk, c * 16, DIM, lane);
          acc[c] = wmma_f32(a, bf, acc[c]);
        }
      }
      int rbase = it * 16 + ((lane >> 4) << 3);
#pragma unroll
      for (int c = 0; c < 4; ++c) {
        int col = c * 16 + (lane & 15);
        float partial = 0.0f;
#pragma unroll
        for (int r = 0; r < 8; ++r) {
          int i = rbase + r;
          if (i < NF) partial += acc[c][r] * sC[(b * NF + i) * DIM + col];
        }
        partial += __shfl_xor(partial, 16, 32);  // fold lane halves (wave32)
        if (lane < 16) atomicAdd(&sOut[b * DIM + col], partial);
      }
    }
    __syncthreads();

    // Drain sOut -> global.
    for (int i = tid; i < BT * DIM; i += 256) {
      int b = i >> 6, dq = i & 63;
      Out[((size_t)(b0 + b) * NF + n) * DIM + dq] = sOut[i];
    }
    __syncthreads();
  }
}

// ---- host launch -----------------------------------------------------------
extern "C" void kernel_launch(void* const* d_in, const int* in_sizes, int n_in,
                              void* d_out, int out_size, void* d_ws,
                              size_t ws_size, hipStream_t stream) {
  const float* inputs = (const float*)d_in[0];   // [B, F, D]
  const float* W      = (const float*)d_in[1];   // [L, N, D, D]
  const float* alpha  = (const float*)d_in[2];   // [L, F, N, N]
  const float* h      = (const float*)d_in[3];   // [L, N, D, 1]
  float* out = (float*)d_out;

  float* ws   = (float*)d_ws;
  float* Mh   = ws;                                   // L*N*D*D   = 491520 f
  float* AlT  = Mh  + LAYERS * NF * DIM * DIM;        // L*N*48*40 = 230400 f
  float* buf0 = AlT + LAYERS * NF * NIPAD * NF;       // B*N*D     = 5242880 f
  float* buf1 = buf0 + (size_t)BATCH * NF * DIM;      // B*N*D

  {
    int t1 = LAYERS * NF * DIM * DIM;
    prep_mh_kernel<<<(t1 + 255) / 256, 256, 0, stream>>>(W, h, Mh, t1);
    int t2 = LAYERS * NF * NIPAD * NF;
    prep_alt_kernel<<<(t2 + 255) / 256, 256, 0, stream>>>(alpha, AlT, t2);
  }

  size_t smem = (size_t)SMEM_FLOATS * sizeof(float);  // ~266 KB of 320 KB LDS
  dim3 grid(BATCH / BT), block(256);

  const size_t mh_l  = (size_t)NF * DIM * DIM;
  const size_t alt_l = (size_t)NF * NIPAD * NF;
  // layer 0: Bi == inputs
  gin_layer_kernel<<<grid, block, smem, stream>>>(inputs, inputs, Mh, AlT, buf0);
  // layer 1
  gin_layer_kernel<<<grid, block, smem, stream>>>(inputs, buf0, Mh + mh_l,
                                                  AlT + alt_l, buf1);
  // layer 2 -> output
  gin_layer_kernel<<<grid, block, smem, stream>>>(inputs, buf1, Mh + 2 * mh_l,
                                                  AlT + 2 * alt_l, out);
}